// GCN_7361573945713
// MI455X (gfx1250) — compile-verified
//
#include <hip/hip_runtime.h>
#include <hip/hip_bf16.h>

// GCN on MI455X (gfx1250, wave32).
//  - GEMMs via V_WMMA_F32_16X16X4_F32 (fp32-exact, GEMM is <5% of roofline time)
//  - Propagation: wave-per-edge gather + global_atomic_add_f32 scatter; the
//    51 MB feature matrices are L2-resident (192 MB L2), edge list streams HBM.

typedef __attribute__((ext_vector_type(2))) float v2f;
typedef __attribute__((ext_vector_type(8))) float v8f;

#define IN_F 128

// ---------------------------------------------------------------- degrees ---
__global__ __launch_bounds__(256) void gcn_deg(const long long* __restrict__ dstA,
                                               long long E, int N,
                                               float* __restrict__ deg) {
  long long i = (long long)blockIdx.x * blockDim.x + threadIdx.x;
  long long tot = E + (long long)N;
  if (i >= tot) return;
  int d = (i < E) ? (int)dstA[i] : (int)(i - E);
  unsafeAtomicAdd(&deg[d], 1.0f);  // no-return global_atomic_add_f32
}

__global__ __launch_bounds__(256) void gcn_rsqrt(float* __restrict__ deg, int N) {
  int i = blockIdx.x * blockDim.x + threadIdx.x;
  if (i >= N) return;
  float d = deg[i];
  deg[i] = (d > 0.0f) ? rsqrtf(d) : 0.0f;  // deg>=1 (self-loops) in practice
}

// ------------------------------------------------------------- WMMA GEMM ---
// Y[N, OUTF] = X[N, IN_F] @ W[OUTF, IN_F]^T
// One wave computes one 16x16 tile of Y; K swept 4 at a time with
// v_wmma_f32_16x16x4_f32.  Fragment layouts per CDNA5 ISA 7.12.2:
//  A (16x4 f32): lanes 0-15 hold M=lane, K={0,1}; lanes 16-31 hold K={2,3}
//  B (4x16 f32): symmetric — half-wave selects K pair, VGPR indexes within
//  C/D (16x16 f32): VGPR r -> row (r + 8*half), col = lane&15
template <int OUTF>
__global__ __launch_bounds__(128) void gcn_gemm_wmma(
    const float* __restrict__ X, const float* __restrict__ W,
    float* __restrict__ Y, int N) {
  const int lane = threadIdx.x & 31;
  const int wave = threadIdx.x >> 5;
  const int tile = blockIdx.x * 4 + wave;
  const int colTiles = OUTF / 16;
  const int numTiles = (N / 16) * colTiles;
  if (tile >= numTiles) return;  // uniform per wave: EXEC stays all-1s

  const int row0 = (tile / colTiles) * 16;
  const int col0 = (tile % colTiles) * 16;
  const int half = lane >> 4;   // which half-wave
  const int sub  = lane & 15;

  const float* __restrict__ xrow = X + (size_t)(row0 + sub) * IN_F;
  const float* __restrict__ wrow = W + (size_t)(col0 + sub) * IN_F;

  v8f c = {};
#pragma unroll
  for (int k0 = 0; k0 < IN_F; k0 += 4) {
    v2f a = {xrow[k0 + 2 * half + 0], xrow[k0 + 2 * half + 1]};
    v2f b = {wrow[k0 + 2 * half + 0], wrow[k0 + 2 * half + 1]};
    // 8 args: (neg_a, A, neg_b, B, c_mod, C, reuse_a, reuse_b)
    c = __builtin_amdgcn_wmma_f32_16x16x4_f32(false, a, false, b, (short)0, c,
                                              false, false);
  }

  float* __restrict__ yp = Y + (size_t)row0 * OUTF + col0 + sub;
#pragma unroll
  for (int r = 0; r < 8; ++r)
    yp[(size_t)(r + 8 * half) * OUTF] = c[r];
}

// ------------------------------------------------------------- propagate ---
// out[dst] += dinv[src]*dinv[dst] * y[src]   (one wave32 per edge)
template <int F>
__global__ __launch_bounds__(256) void gcn_propagate(
    const float* __restrict__ Y, const long long* __restrict__ srcA,
    const long long* __restrict__ dstA, const float* __restrict__ dinv,
    float* __restrict__ Out, long long E, int N) {
  constexpr int V = F / 32;  // floats per lane (4 for F=128, 2 for F=64)
  const int lane = threadIdx.x & 31;
  long long e = (long long)blockIdx.x * (blockDim.x >> 5) + (threadIdx.x >> 5);
  const long long tot = E + (long long)N;
  if (e >= tot) return;

  int s, d;
  if (e < E) {
    s = (int)srcA[e];
    d = (int)dstA[e];
  } else {
    s = d = (int)(e - E);  // self-loop
  }
  const float nrm = dinv[s] * dinv[d];

  const float* __restrict__ yp = Y + (size_t)s * F + lane * V;
  float* __restrict__ op = Out + (size_t)d * F + lane * V;

  float vals[V];
#pragma unroll
  for (int v = 0; v < V; ++v) vals[v] = yp[v];  // merges to global_load_b128
#pragma unroll
  for (int v = 0; v < V; ++v) unsafeAtomicAdd(op + v, nrm * vals[v]);
}

// ----------------------------------------------------------------- driver ---
extern "C" void kernel_launch(void* const* d_in, const int* in_sizes, int n_in,
                              void* d_out, int out_size, void* d_ws,
                              size_t ws_size, hipStream_t stream) {
  (void)n_in; (void)out_size; (void)ws_size;

  const float*     x   = (const float*)d_in[0];
  const long long* ei  = (const long long*)d_in[1];  // int64 per reference
  const float*     W1  = (const float*)d_in[2];      // [128,128]
  const float*     W2  = (const float*)d_in[3];      // [64,128]

  const int       N = in_sizes[0] / IN_F;            // 100000 (divisible by 16)
  const long long E = (long long)in_sizes[1] / 2;    // 1600000
  const long long* srcA = ei;
  const long long* dstA = ei + E;
  const long long tot = E + (long long)N;

  // workspace layout: dinv[N] | bufA[N*128] | bufB[N*128]
  float* dinv = (float*)d_ws;
  size_t off  = ((size_t)N + 255) & ~(size_t)255;
  float* bufA = (float*)d_ws + off;
  float* bufB = bufA + (size_t)N * IN_F;

  const int degBlocks  = (int)((tot + 255) / 256);
  const int propBlocks = (int)((tot + 7) / 8);            // 8 waves / block
  const int gemmBlocks128 = ((N / 16) * (128 / 16) + 3) / 4;
  const int gemmBlocks64  = ((N / 16) * (64 / 16) + 3) / 4;
  const size_t featBytes = (size_t)N * IN_F * sizeof(float);

  // symmetric normalization coefficients
  hipMemsetAsync(dinv, 0, (size_t)N * sizeof(float), stream);
  gcn_deg<<<degBlocks, 256, 0, stream>>>(dstA, E, N, dinv);
  gcn_rsqrt<<<(N + 255) / 256, 256, 0, stream>>>(dinv, N);

  // layer 1: h = P(x @ W1^T)
  gcn_gemm_wmma<128><<<gemmBlocks128, 128, 0, stream>>>(x, W1, bufA, N);
  hipMemsetAsync(bufB, 0, featBytes, stream);
  gcn_propagate<128><<<propBlocks, 256, 0, stream>>>(bufA, srcA, dstA, dinv,
                                                     bufB, E, N);

  // layer 2: h = P(h @ W1^T)
  gcn_gemm_wmma<128><<<gemmBlocks128, 128, 0, stream>>>(bufB, W1, bufA, N);
  hipMemsetAsync(bufB, 0, featBytes, stream);
  gcn_propagate<128><<<propBlocks, 256, 0, stream>>>(bufA, srcA, dstA, dinv,
                                                     bufB, E, N);

  // layer 3: out = P(h @ W2^T)
  gcn_gemm_wmma<64><<<gemmBlocks64, 128, 0, stream>>>(bufB, W2, bufA, N);
  hipMemsetAsync(d_out, 0, (size_t)N * 64 * sizeof(float), stream);
  gcn_propagate<64><<<propBlocks, 256, 0, stream>>>(bufA, srcA, dstA, dinv,
                                                    (float*)d_out, E, N);
}